// RegularizedAMLGNN_13726715478162
// MI455X (gfx1250) — compile-verified
//
#include <hip/hip_runtime.h>

typedef __attribute__((ext_vector_type(2))) float v2f;
typedef __attribute__((ext_vector_type(8))) float v8f;

__device__ __forceinline__ v8f wmma_k4(v2f a, v2f b, v8f c) {
  // V_WMMA_F32_16X16X4_F32 : D = A(16x4) * B(4x16) + C(16x16), fp32 exact
  return __builtin_amdgcn_wmma_f32_16x16x4_f32(false, a, false, b, (short)0, c,
                                               false, false);
}

// branch-free jnp.nan_to_num: NaN->0 (first), +/-Inf -> +/-FLT_MAX
__device__ __forceinline__ float fixnum(float v) {
  v = (v != v) ? 0.0f : v;                        // v_cmp + v_cndmask
  v = fminf(fmaxf(v, -3.4028234663852886e38f),    // v_max_f32
            3.4028234663852886e38f);              // v_min_f32
  return v;
}

__device__ __forceinline__ v2f ld2(const float* __restrict__ p) {
  const float2 t = *reinterpret_cast<const float2*>(p);
  v2f r; r.x = t.x; r.y = t.y;
  return r;
}

// ---------------------------------------------------------------- utilities
__global__ void zero_kernel(float* __restrict__ p, long n) {
  long i = (long)blockIdx.x * blockDim.x + threadIdx.x;
  if (i < n) p[i] = 0.0f;
}

__global__ void degree_kernel(const int* __restrict__ ei, float* __restrict__ deg,
                              long E) {
  long e = (long)blockIdx.x * blockDim.x + threadIdx.x;
  if (e < E) atomicAdd(&deg[ei[E + e]], 1.0f);
}

__global__ void dinv_kernel(float* __restrict__ deg, long n) {
  long i = (long)blockIdx.x * blockDim.x + threadIdx.x;
  if (i < n) deg[i] = rsqrtf(deg[i] + 1.0f);
}

// ------------------------------------------------------- node-feature GEMM
// Y[M x (NT*16)] = X[M x K] @ W[K x (NT*16)]     (one 16-row tile per wave)
// LDS weights are K-pair interleaved: sW[(k&~1)*NC + 2*col + (k&1)] = W[k][col]
// so each B operand fetch is one aligned ds_load_b64 into an even VGPR pair.
template <int K, int NT, bool NANFIX>
__global__ void gemm_wmma(const float* __restrict__ X, const float* __restrict__ W,
                          float* __restrict__ Y, int M) {
  constexpr int NC = NT * 16;
  constexpr int LOG = (NC == 64) ? 6 : ((NC == 32) ? 5 : 4);
  __shared__ alignas(16) float sW[K * NC];
  for (int i = threadIdx.x; i < K * NC; i += blockDim.x) {
    const int k = i >> LOG;
    const int c = i & (NC - 1);
    sW[(k & ~1) * NC + 2 * c + (k & 1)] = W[i];
  }
  __syncthreads();

  const int lane = threadIdx.x & 31;
  const int wave = threadIdx.x >> 5;
  const int lq = lane & 15;
  const int half = lane >> 4;
  const int wpb = blockDim.x >> 5;

  for (long tile = (long)blockIdx.x * wpb + wave; tile * 16 < M;
       tile += (long)gridDim.x * wpb) {
    // clamp: out-of-range rows read row M-1; their D rows are never stored
    const int row = min((int)(tile * 16) + lq, M - 1);
    const float* xr = X + (size_t)row * K;

    v8f acc[NT];
#pragma unroll
    for (int t = 0; t < NT; ++t) acc[t] = (v8f){0, 0, 0, 0, 0, 0, 0, 0};

#pragma unroll
    for (int k = 0; k < K; k += 4) {
      const int k0 = k + 2 * half;
      v2f a = ld2(xr + k0);  // one b64 load: elements k0, k0+1
      if (NANFIX) { a.x = fixnum(a.x); a.y = fixnum(a.y); }
#pragma unroll
      for (int t = 0; t < NT; ++t) {
        const int col = t * 16 + lq;
        v2f b = ld2(&sW[k0 * NC + 2 * col]);  // single ds_load_b64
        acc[t] = wmma_k4(a, b, acc[t]);
      }
    }

    const bool fullTile = ((int)(tile * 16) + 16) <= M;
#pragma unroll
    for (int t = 0; t < NT; ++t) {
#pragma unroll
      for (int j = 0; j < 8; ++j) {
        const int r = (int)(tile * 16) + j + 8 * half;
        if (fullTile || r < M)
          Y[(size_t)r * NC + t * 16 + lq] = acc[t][j];
      }
    }
  }
}

// ------------------------------------------------------ edge scatter (agg)
template <int H>
__global__ void scatter_kernel(const int* __restrict__ ei,
                               const float* __restrict__ dinv,
                               const float* __restrict__ xw,
                               float* __restrict__ agg, long E) {
  constexpr int CH = H / 4;
  constexpr int SH = (CH == 16) ? 4 : ((CH == 8) ? 3 : 2);
  long idx = (long)blockIdx.x * blockDim.x + threadIdx.x;
  if (idx >= E * CH) return;
  long e = idx >> SH;
  int c = (int)(idx & (CH - 1));
  int s = ei[e], d = ei[E + e];
  float nrm = dinv[s] * dinv[d];
  const float4 v = *(const float4*)(xw + (size_t)s * H + c * 4);
  float* ap = agg + (size_t)d * H + c * 4;
  atomicAdd(ap + 0, v.x * nrm);
  atomicAdd(ap + 1, v.y * nrm);
  atomicAdd(ap + 2, v.z * nrm);
  atomicAdd(ap + 3, v.w * nrm);
}

// ------------------------------------ self-loop + bias + BatchNorm + ReLU
template <int H>
__global__ void finalize_kernel(const float* __restrict__ xw,
                                const float* __restrict__ agg,
                                const float* __restrict__ dinv,
                                const float* __restrict__ b,
                                const float* __restrict__ g,
                                const float* __restrict__ be,
                                const float* __restrict__ m,
                                const float* __restrict__ v,
                                float* __restrict__ h, long Nn) {
  constexpr int SH = (H == 64) ? 6 : ((H == 32) ? 5 : 4);
  long idx = (long)blockIdx.x * blockDim.x + threadIdx.x;
  if (idx >= Nn * H) return;
  long i = idx >> SH;
  int f = (int)(idx & (H - 1));
  float di = dinv[i];
  float val = agg[idx] + xw[idx] * di * di + b[f];
  val = (val - m[f]) * rsqrtf(v[f] + 1e-5f) * g[f] + be[f];
  h[idx] = fmaxf(val, 0.0f);
}

// --------------------------------------------------------- edge MLP (WMMA)
// ef = [h3[src](16) | h3[dst](16) | edge_attr(8)]  ->  40 -> 32 -> 16 -> 2
// All weight LDS arrays use the K-pair interleaved layout (see gemm_wmma).
__global__ void edge_mlp_kernel(const float* __restrict__ h3,
                                const int* __restrict__ ei,
                                const float* __restrict__ ea,
                                const float* __restrict__ Wc1,
                                const float* __restrict__ bc1,
                                const float* __restrict__ Wc2,
                                const float* __restrict__ bc2,
                                const float* __restrict__ Wc3,
                                const float* __restrict__ bc3,
                                float* __restrict__ out, long E) {
  constexpr int STR = 34;  // even stride: b64-aligned, conflict-free (17i mod 32)
  __shared__ alignas(16) float sW1[40 * 32];
  __shared__ alignas(16) float sW2[32 * 16];
  __shared__ alignas(16) float sW3[16 * 16];
  __shared__ float sB1[32], sB2[16], sB3[16];
  __shared__ alignas(16) float stage[8][16 * STR];  // per-wave C->A relayout

  for (int i = threadIdx.x; i < 40 * 32; i += blockDim.x) {
    const int k = i >> 5, c = i & 31;
    sW1[(k & ~1) * 32 + 2 * c + (k & 1)] = Wc1[i];
  }
  for (int i = threadIdx.x; i < 32 * 16; i += blockDim.x) {
    const int k = i >> 4, c = i & 15;
    sW2[(k & ~1) * 16 + 2 * c + (k & 1)] = Wc2[i];
  }
  for (int i = threadIdx.x; i < 16 * 16; i += blockDim.x) {
    const int k = i >> 4, c = i & 15;
    sW3[(k & ~1) * 16 + 2 * c + (k & 1)] = (c < 2) ? Wc3[k * 2 + c] : 0.0f;
  }
  for (int i = threadIdx.x; i < 32; i += blockDim.x) sB1[i] = bc1[i];
  for (int i = threadIdx.x; i < 16; i += blockDim.x) sB2[i] = bc2[i];
  for (int i = threadIdx.x; i < 16; i += blockDim.x)
    sB3[i] = (i < 2) ? bc3[i] : 0.0f;
  __syncthreads();

  const int lane = threadIdx.x & 31;
  const int wave = threadIdx.x >> 5;
  const int lq = lane & 15;
  const int half = lane >> 4;
  float* st = &stage[wave][0];

  const long numTiles = (E + 15) >> 4;
  for (long tile = (long)blockIdx.x * 8 + wave; tile < numTiles;
       tile += (long)gridDim.x * 8) {
    // clamp: OOB edges read edge E-1; their D rows are never stored
    const long edge = min(tile * 16 + lq, E - 1);
    const int s = ei[edge];
    const int d = ei[E + edge];
    const float* srow = h3 + (size_t)s * 16;
    const float* drow = h3 + (size_t)d * 16;
    const float* earow = ea + edge * 8;

    // ---- layer 1: [16x40] @ [40x32]
    v8f a0c = (v8f){0, 0, 0, 0, 0, 0, 0, 0};
    v8f a1c = (v8f){0, 0, 0, 0, 0, 0, 0, 0};
#pragma unroll
    for (int k = 0; k < 40; k += 4) {
      const int k0 = k + 2 * half;  // pair (k0,k0+1) never straddles a segment
      v2f a;
      if (k < 16) {
        a = ld2(srow + k0);
      } else if (k < 32) {
        a = ld2(drow + (k0 - 16));
      } else {
        a = ld2(earow + (k0 - 32));
        a.x = fixnum(a.x);
        a.y = fixnum(a.y);
      }
      v2f b0 = ld2(&sW1[k0 * 32 + 2 * lq]);         // ds_load_b64
      v2f b1 = ld2(&sW1[k0 * 32 + 2 * (16 + lq)]);  // ds_load_b64
      a0c = wmma_k4(a, b0, a0c);
      a1c = wmma_k4(a, b1, a1c);
    }
    // bias + relu, write z1[16x32] to staging in row-major
    __builtin_amdgcn_wave_barrier();
#pragma unroll
    for (int j = 0; j < 8; ++j) {
      const int r = j + 8 * half;
      st[r * STR + lq]      = fmaxf(a0c[j] + sB1[lq], 0.0f);
      st[r * STR + 16 + lq] = fmaxf(a1c[j] + sB1[16 + lq], 0.0f);
    }
    __builtin_amdgcn_wave_barrier();

    // ---- layer 2: [16x32] @ [32x16]
    v8f a2c = (v8f){0, 0, 0, 0, 0, 0, 0, 0};
#pragma unroll
    for (int k = 0; k < 32; k += 4) {
      const int k0 = k + 2 * half;
      v2f a = ld2(&st[lq * STR + k0]);          // aligned ds_load_b64
      v2f b = ld2(&sW2[k0 * 16 + 2 * lq]);      // ds_load_b64
      a2c = wmma_k4(a, b, a2c);
    }
    __builtin_amdgcn_wave_barrier();
#pragma unroll
    for (int j = 0; j < 8; ++j) {
      const int r = j + 8 * half;
      st[r * STR + lq] = fmaxf(a2c[j] + sB2[lq], 0.0f);
    }
    __builtin_amdgcn_wave_barrier();

    // ---- layer 3: [16x16] @ [16x16] (only 2 output cols are real)
    v8f a3c = (v8f){0, 0, 0, 0, 0, 0, 0, 0};
#pragma unroll
    for (int k = 0; k < 16; k += 4) {
      const int k0 = k + 2 * half;
      v2f a = ld2(&st[lq * STR + k0]);
      v2f b = ld2(&sW3[k0 * 16 + 2 * lq]);
      a3c = wmma_k4(a, b, a3c);
    }
    if (lq < 2) {
#pragma unroll
      for (int j = 0; j < 8; ++j) {
        const long e2 = tile * 16 + j + 8 * half;
        if (e2 < E) out[e2 * 2 + lq] = fixnum(a3c[j] + sB3[lq]);
      }
    }
    __builtin_amdgcn_wave_barrier();
  }
}

// --------------------------------------------------------------- launcher
extern "C" void kernel_launch(void* const* d_in, const int* in_sizes, int n_in,
                              void* d_out, int out_size, void* d_ws,
                              size_t ws_size, hipStream_t stream) {
  const float* x   = (const float*)d_in[0];
  const int*   ei  = (const int*)d_in[1];
  const float* ea  = (const float*)d_in[2];
  const float* W1  = (const float*)d_in[3];
  const float* b1  = (const float*)d_in[4];
  const float* W2  = (const float*)d_in[5];
  const float* b2  = (const float*)d_in[6];
  const float* W3  = (const float*)d_in[7];
  const float* b3  = (const float*)d_in[8];
  const float* g1  = (const float*)d_in[9];
  const float* be1 = (const float*)d_in[10];
  const float* m1  = (const float*)d_in[11];
  const float* v1  = (const float*)d_in[12];
  const float* g2  = (const float*)d_in[13];
  const float* be2 = (const float*)d_in[14];
  const float* m2  = (const float*)d_in[15];
  const float* v2  = (const float*)d_in[16];
  const float* g3  = (const float*)d_in[17];
  const float* be3 = (const float*)d_in[18];
  const float* m3  = (const float*)d_in[19];
  const float* v3  = (const float*)d_in[20];
  const float* Wc1 = (const float*)d_in[21];
  const float* bc1 = (const float*)d_in[22];
  const float* Wc2 = (const float*)d_in[23];
  const float* bc2 = (const float*)d_in[24];
  const float* Wc3 = (const float*)d_in[25];
  const float* bc3 = (const float*)d_in[26];

  const long Nn = in_sizes[0] / 128;  // 100000
  const long E  = in_sizes[2] / 8;    // 3200000

  float* ws   = (float*)d_ws;
  float* dinv = ws;                 // Nn
  float* xw   = dinv + Nn;          // Nn*64
  float* agg  = xw + Nn * 64;       // Nn*64
  float* h1   = agg + Nn * 64;      // Nn*64
  float* h2   = h1 + Nn * 64;       // Nn*32
  float* h3   = h2 + Nn * 32;       // Nn*16

  const int TB = 256;
  auto blk = [](long n) { return (int)((n + 255) / 256); };
  const int rowTiles = (int)((Nn + 15) / 16);
  const int gemmBlocks = (rowTiles + 7) / 8;

  // degree -> dinv = rsqrt(deg+1)
  zero_kernel<<<blk(Nn), TB, 0, stream>>>(dinv, Nn);
  degree_kernel<<<blk(E), TB, 0, stream>>>(ei, dinv, E);
  dinv_kernel<<<blk(Nn), TB, 0, stream>>>(dinv, Nn);

  // ---- GCN layer 1 (128 -> 64)
  gemm_wmma<128, 4, true><<<gemmBlocks, TB, 0, stream>>>(x, W1, xw, (int)Nn);
  zero_kernel<<<blk(Nn * 64), TB, 0, stream>>>(agg, Nn * 64);
  scatter_kernel<64><<<blk(E * 16), TB, 0, stream>>>(ei, dinv, xw, agg, E);
  finalize_kernel<64><<<blk(Nn * 64), TB, 0, stream>>>(xw, agg, dinv, b1, g1,
                                                       be1, m1, v1, h1, Nn);
  // ---- GCN layer 2 (64 -> 32)
  gemm_wmma<64, 2, false><<<gemmBlocks, TB, 0, stream>>>(h1, W2, xw, (int)Nn);
  zero_kernel<<<blk(Nn * 32), TB, 0, stream>>>(agg, Nn * 32);
  scatter_kernel<32><<<blk(E * 8), TB, 0, stream>>>(ei, dinv, xw, agg, E);
  finalize_kernel<32><<<blk(Nn * 32), TB, 0, stream>>>(xw, agg, dinv, b2, g2,
                                                       be2, m2, v2, h2, Nn);
  // ---- GCN layer 3 (32 -> 16)
  gemm_wmma<32, 1, false><<<gemmBlocks, TB, 0, stream>>>(h2, W3, xw, (int)Nn);
  zero_kernel<<<blk(Nn * 16), TB, 0, stream>>>(agg, Nn * 16);
  scatter_kernel<16><<<blk(E * 4), TB, 0, stream>>>(ei, dinv, xw, agg, E);
  finalize_kernel<16><<<blk(Nn * 16), TB, 0, stream>>>(xw, agg, dinv, b3, g3,
                                                       be3, m3, v3, h3, Nn);

  // ---- edge classifier MLP on WMMA
  const long edgeTiles = (E + 15) / 16;
  const int mlpBlocks = (int)((edgeTiles + 7) / 8);
  edge_mlp_kernel<<<mlpBlocks, TB, 0, stream>>>(h3, ei, ea, Wc1, bc1, Wc2, bc2,
                                                Wc3, bc3, (float*)d_out, E);
}